// GraphCON_6253472383695
// MI455X (gfx1250) — compile-verified
//
#include <hip/hip_runtime.h>
#include <math.h>

typedef __attribute__((ext_vector_type(2))) float v2f;
typedef __attribute__((ext_vector_type(8))) float v8f;

#define HID 64
#define SLOPE 0.01f
#define DT 1.0f
#define ALPHA 1.0f
#define GAMMA 1.0f

// ---------------- CSR construction ----------------

__global__ void zero_i32(int* __restrict__ p, int n) {
    int i = blockIdx.x * blockDim.x + threadIdx.x;
    if (i < n) p[i] = 0;
}

__global__ void count_k(const int* __restrict__ dst, int* __restrict__ cnt, int E) {
    int e = blockIdx.x * blockDim.x + threadIdx.x;
    if (e < E) atomicAdd(&cnt[dst[e]], 1);
}

__global__ void dinv_k(const int* __restrict__ cnt, float* __restrict__ dinv, int n) {
    int i = blockIdx.x * blockDim.x + threadIdx.x;
    if (i < n) dinv[i] = rsqrtf((float)cnt[i] + 1.0f);  // +1 self loop
}

// single-block exclusive scan over n entries (runs once per call, n=100000)
__global__ void scan_k(const int* __restrict__ cnt, int* __restrict__ off, int n) {
    __shared__ int sh[1024];
    __shared__ int carry_s;
    int tid = threadIdx.x;
    if (tid == 0) carry_s = 0;
    __syncthreads();
    for (int base = 0; base < n; base += 1024) {
        int i = base + tid;
        int v = (i < n) ? cnt[i] : 0;
        sh[tid] = v;
        __syncthreads();
        for (int d = 1; d < 1024; d <<= 1) {
            int t = (tid >= d) ? sh[tid - d] : 0;
            __syncthreads();
            sh[tid] += t;
            __syncthreads();
        }
        int incl = sh[tid];
        int carry = carry_s;
        if (i < n) off[i] = carry + incl - v;
        __syncthreads();
        if (tid == 1023) carry_s = carry + incl;
        __syncthreads();
    }
    if (threadIdx.x == 0) off[n] = carry_s;
}

__global__ void fill_k(const int* __restrict__ dst, const int* __restrict__ off,
                       int* __restrict__ cursor, int* __restrict__ eid, int E) {
    int e = blockIdx.x * blockDim.x + threadIdx.x;
    if (e >= E) return;
    int d = dst[e];
    int p = off[d] + atomicAdd(&cursor[d], 1);
    eid[p] = e;
}

// per-node insertion sort of edge ids -> deterministic summation order,
// then materialize src index + normalized weight
__global__ void sort_finalize_k(const int* __restrict__ srcA, const int* __restrict__ off,
                                int* __restrict__ eid, int* __restrict__ csr_src,
                                float* __restrict__ csr_w, const float* __restrict__ dinv, int n) {
    int i = blockIdx.x * blockDim.x + threadIdx.x;
    if (i >= n) return;
    int beg = off[i], end = off[i + 1];
    for (int a = beg + 1; a < end; ++a) {
        int v = eid[a];
        int b = a - 1;
        while (b >= beg && eid[b] > v) { eid[b + 1] = eid[b]; --b; }
        eid[b + 1] = v;
    }
    float di = dinv[i];
    for (int p = beg; p < end; ++p) {
        int e = eid[p];
        int s = srcA[e];
        csr_src[p] = s;
        csr_w[p] = dinv[s] * di;
    }
}

// ---------------- WMMA GEMM: C[M,NC] = A[M,KD] @ B[KD,NC] (+epilogue) ----------------
// MODE 0: plain store. MODE 1: tanh(acc+bias) -> C and C2. MODE 2: leakyrelu(acc+bias) -> C.

template <int KD, int NC, int MODE>
__global__ __launch_bounds__(256) void gemm_wmma(const float* __restrict__ A,
                                                 const float* __restrict__ B,
                                                 const float* __restrict__ bias,
                                                 float* __restrict__ C,
                                                 float* __restrict__ C2, int M) {
    constexpr int COL_TILES = NC / 16;            // waves across N
    constexpr int ROWS = (8 / COL_TILES) * 16;    // rows per 256-thread block
    int wave = threadIdx.x >> 5;
    int lane = threadIdx.x & 31;
    int half = lane >> 4;
    int l = lane & 15;
    int m0 = blockIdx.x * ROWS + (wave / COL_TILES) * 16;
    int n0 = (wave % COL_TILES) * 16;

    int arow = m0 + l;
    if (arow >= M) arow = M - 1;  // clamp: keeps EXEC uniform, garbage rows never stored
    const float* Arow = A + (size_t)arow * KD;

    v8f acc = {};
#pragma unroll
    for (int kt = 0; kt < KD / 4; ++kt) {
        int k0 = kt * 4 + half * 2;
        v2f a;
        a.x = Arow[k0];
        a.y = Arow[k0 + 1];
        v2f b;
        b.x = B[(size_t)k0 * NC + n0 + l];
        b.y = B[(size_t)(k0 + 1) * NC + n0 + l];
        acc = __builtin_amdgcn_wmma_f32_16x16x4_f32(false, a, false, b, (short)0, acc,
                                                    false, false);
    }

    int crow0 = m0 + half * 8;
    int ccol = n0 + l;
    float bv = (MODE != 0) ? bias[ccol] : 0.0f;
#pragma unroll
    for (int v = 0; v < 8; ++v) {
        int r = crow0 + v;
        if (r < M) {
            float val = acc[v];
            if (MODE == 1) {
                val = tanhf(val + bv);
                C2[(size_t)r * NC + ccol] = val;
            } else if (MODE == 2) {
                val += bv;
                val = val > 0.0f ? val : SLOPE * val;
            }
            C[(size_t)r * NC + ccol] = val;
        }
    }
}

// ---------------- fused aggregation + GraphCON update ----------------
// 64 threads per node (feature per thread), coalesced gathers of H[src].

__global__ __launch_bounds__(256) void agg_update_k(
    const float* __restrict__ H, const float* __restrict__ R, float* __restrict__ X,
    float* __restrict__ Y, const int* __restrict__ off, const int* __restrict__ csr_src,
    const float* __restrict__ csr_w, const float* __restrict__ dinv,
    const float* __restrict__ b_c, const float* __restrict__ b_r, int n) {
    int gid = blockIdx.x * blockDim.x + threadIdx.x;
    int node = gid >> 6;
    int f = gid & 63;
    if (node >= n) return;
    float di = dinv[node];
    size_t base = (size_t)node * HID + f;
    float acc = di * di * H[base];  // self loop
    int beg = off[node], end = off[node + 1];
    for (int e = beg; e < end; ++e) {
        acc = fmaf(csr_w[e], H[(size_t)csr_src[e] * HID + f], acc);
    }
    float g = acc + b_c[f] - R[base] - b_r[f];
    float t = tanhf(g);
    float x = X[base], y = Y[base];
    float yn = y + DT * (t - ALPHA * y - GAMMA * x);
    float xn = x + DT * yn;
    X[base] = xn;
    Y[base] = yn;
}

// ---------------- host-side orchestration ----------------

static inline char* carve(char*& p, size_t bytes) {
    char* r = p;
    p += (bytes + 255) & ~(size_t)255;
    return r;
}

extern "C" void kernel_launch(void* const* d_in, const int* in_sizes, int n_in, void* d_out,
                              int out_size, void* d_ws, size_t ws_size, hipStream_t stream) {
    const float* x = (const float*)d_in[0];
    const int* eidx = (const int*)d_in[1];
    const float* W_emb = (const float*)d_in[2];
    const float* b_emb = (const float*)d_in[3];
    const float* W_c = (const float*)d_in[4];
    const float* b_c = (const float*)d_in[5];
    const float* W_r = (const float*)d_in[6];
    const float* b_r = (const float*)d_in[7];
    const float* W_1 = (const float*)d_in[8];
    const float* b_1 = (const float*)d_in[9];
    const float* W_2 = (const float*)d_in[10];
    const float* b_2 = (const float*)d_in[11];

    const int N = in_sizes[0] / HID;  // 100000
    const int E = in_sizes[1] / 2;    // 1600000
    const int* srcA = eidx;
    const int* dstA = eidx + E;

    char* p = (char*)d_ws;
    int* cnt = (int*)carve(p, (size_t)N * 4);
    int* cursor = (int*)carve(p, (size_t)N * 4);
    int* off = (int*)carve(p, (size_t)(N + 1) * 4);
    float* dinv = (float*)carve(p, (size_t)N * 4);
    int* eid = (int*)carve(p, (size_t)E * 4);
    int* csr_src = (int*)carve(p, (size_t)E * 4);
    float* csr_w = (float*)carve(p, (size_t)E * 4);
    float* X = (float*)carve(p, (size_t)N * HID * 4);
    float* Y = (float*)carve(p, (size_t)N * HID * 4);
    float* H = (float*)carve(p, (size_t)N * HID * 4);
    float* R = (float*)carve(p, (size_t)N * HID * 4);
    float* H1 = (float*)carve(p, (size_t)N * 32 * 4);

    const int T = 256;
    int gN = (N + T - 1) / T;
    int gE = (E + T - 1) / T;

    // CSR build (once per call; deterministic via per-node sort)
    zero_i32<<<gN, T, 0, stream>>>(cnt, N);
    zero_i32<<<gN, T, 0, stream>>>(cursor, N);
    count_k<<<gE, T, 0, stream>>>(dstA, cnt, E);
    dinv_k<<<gN, T, 0, stream>>>(cnt, dinv, N);
    scan_k<<<1, 1024, 0, stream>>>(cnt, off, N);
    fill_k<<<gE, T, 0, stream>>>(dstA, off, cursor, eid, E);
    sort_finalize_k<<<gN, T, 0, stream>>>(srcA, off, eid, csr_src, csr_w, dinv, N);

    // Embedding: X = Y = tanh(x @ W_emb + b_emb)
    int g64 = (N + 31) / 32;  // 32 rows/block for NC=64
    int g32 = (N + 63) / 64;  // 64 rows/block for NC=32
    gemm_wmma<64, 64, 1><<<g64, 256, 0, stream>>>(x, W_emb, b_emb, X, Y, N);

    // 10 GraphCON layers
    int gAgg = ((size_t)N * HID + T - 1) / T;
    for (int it = 0; it < 10; ++it) {
        gemm_wmma<64, 64, 0><<<g64, 256, 0, stream>>>(X, W_c, nullptr, H, nullptr, N);
        gemm_wmma<64, 64, 0><<<g64, 256, 0, stream>>>(H, W_r, nullptr, R, nullptr, N);
        agg_update_k<<<gAgg, T, 0, stream>>>(H, R, X, Y, off, csr_src, csr_w, dinv, b_c, b_r, N);
    }

    // Readout MLP
    gemm_wmma<64, 32, 2><<<g32, 256, 0, stream>>>(X, W_1, b_1, H1, nullptr, N);
    gemm_wmma<32, 32, 2><<<g32, 256, 0, stream>>>(H1, W_2, b_2, (float*)d_out, nullptr, N);
}